// NERStructureAwareLoss_37391985279355
// MI455X (gfx1250) — compile-verified
//
#include <hip/hip_runtime.h>

typedef float v2f __attribute__((ext_vector_type(2)));
typedef float v8f __attribute__((ext_vector_type(8)));

#define C_TAGS   38
#define ROWSTR   41          // LDS row stride in floats (odd -> bank-conflict-free, >= 40 for K-pad)
#define T_SEQ    4096
#define PAD_IDX  0
#define O_IDX    1

// quantities: 0 ce_sum 1 ce_cnt 2 miss_sum 3 miss_cnt 4 fp_sum 5 fp_cnt 6 tr_sum 7 tr_cnt

__global__ __launch_bounds__(256) void ner_loss_main(
    const float* __restrict__ logits,
    const int*   __restrict__ labels,
    const float* __restrict__ validmask,
    float* __restrict__ partials,
    int total)
{
  __shared__ float sl[257 * ROWSTR];     // logits -> probs, 257 tokens (256 + halo)
  __shared__ float sinv[48 * ROWSTR];    // invalid = 1 - valid, padded to 48x40 with zeros
  __shared__ int   slab[257];
  __shared__ float wred[8 * 8];

  const int tid = threadIdx.x;
  const int g0  = blockIdx.x * 256;

  // ---- stage 1: async global->LDS staging of the logits slab (coalesced) ----
  int ntok = total - g0; if (ntok > 257) ntok = 257;
  {
    const float* src = logits + (size_t)g0 * C_TAGS;
    int elems = ntok * C_TAGS;
    for (int e = tid; e < elems; e += 256) {
      int t = e / C_TAGS;
      int c = e - t * C_TAGS;
      unsigned lds_dst = (unsigned)(unsigned long long)(uintptr_t)&sl[t * ROWSTR + c];
      unsigned voff = (unsigned)e * 4u;
      // GVS mode: copies 4 bytes from global address saddr plus offset vaddr to the LDS address in vdst; tracked by ASYNCcnt
      asm volatile("global_load_async_to_lds_b32 %0, %1, %2"
                   :: "v"(lds_dst), "v"(voff), "s"(src)
                   : "memory");
    }
    for (int j = tid; j < 257; j += 256) {
      sl[j * ROWSTR + 38] = 0.f;
      sl[j * ROWSTR + 39] = 0.f;
      sl[j * ROWSTR + 40] = 0.f;
      if (j >= ntok) {
#pragma unroll
        for (int c = 0; c < C_TAGS; ++c) sl[j * ROWSTR + c] = 0.f;
      }
      int g = g0 + j;
      slab[j] = (g < total) ? labels[g] : PAD_IDX;
    }
    for (int e = tid; e < 48 * ROWSTR; e += 256) {
      int r = e / ROWSTR, c = e - r * ROWSTR;
      float v = 0.f;
      if (r < C_TAGS && c < C_TAGS) v = 1.0f - validmask[r * C_TAGS + c];
      sinv[e] = v;
    }
    asm volatile("s_wait_asynccnt 0" ::: "memory");
  }
  __syncthreads();

  // -------- stage 2: in-place softmax + CE / miss / FP partial sums --------
  float ce_s = 0.f, ce_c = 0.f, ms_s = 0.f, ms_c = 0.f, fp_s = 0.f, fp_c = 0.f;
  for (int j = tid; j < 257; j += 256) {
    if (g0 + j < total) {
      float* row = &sl[j * ROWSTR];
      float xs[C_TAGS];
#pragma unroll
      for (int c = 0; c < C_TAGS; ++c) xs[c] = row[c];
      float m = xs[0];
#pragma unroll
      for (int c = 1; c < C_TAGS; ++c) m = fmaxf(m, xs[c]);
      float s = 0.f;
#pragma unroll
      for (int c = 0; c < C_TAGS; ++c) { xs[c] = __expf(xs[c] - m); s += xs[c]; }
      int lab = slab[j];
      float gold_x = row[lab];           // logit of gold label (before overwrite)
      float inv_s = 1.0f / s;
#pragma unroll
      for (int c = 0; c < C_TAGS; ++c) row[c] = xs[c] * inv_s;   // write probs
      if (j < 256 && lab != PAD_IDX) {
        float logs = __logf(s);
        ce_s += -(gold_x - m - logs); ce_c += 1.f;
        float p_o = xs[O_IDX] * inv_s;
        if (lab != O_IDX) { ms_s += -__logf(fmaxf(1.f - p_o, 1e-8f)); ms_c += 1.f; }
        else              { fp_s += -__logf(fmaxf(p_o,       1e-8f)); fp_c += 1.f; }
      }
    }
  }
  __syncthreads();

  // -------- stage 3: transition penalty, 32 pairs per wave (2 groups of 16) via WMMA --------
  float tr_s = 0.f, tr_c = 0.f;
  {
    const int lane = tid & 31;
    const int wv   = tid >> 5;
    const int li   = lane & 15;
    const int half = lane >> 4;

    // hoist the 30 A-operand tiles of the invalid matrix (group-invariant)
    v2f At[30];
#pragma unroll
    for (int mt = 0; mt < 3; ++mt) {
      const int arow = (mt * 16 + li) * ROWSTR;
#pragma unroll
      for (int kc = 0; kc < 10; ++kc) {
        const int c0 = kc * 4 + half * 2;
        v2f a;
        a.x = sinv[arow + c0];
        a.y = sinv[arow + c0 + 1];
        At[mt * 10 + kc] = a;
      }
    }

#pragma unroll
    for (int grp = 0; grp < 2; ++grp) {
      const int base = wv * 32 + grp * 16;     // first prev-token of this 16-pair group
      // W(48x16) = sinv(48x40) x PcurrT(40x16); Pcurr rows are tokens base+1 .. base+16
      v8f W[3];
      const int brow = (base + 1 + li) * ROWSTR;
#pragma unroll
      for (int mt = 0; mt < 3; ++mt) {
        v8f acc = {0.f, 0.f, 0.f, 0.f, 0.f, 0.f, 0.f, 0.f};
#pragma unroll
        for (int kc = 0; kc < 10; ++kc) {
          const int c0 = kc * 4 + half * 2;
          v2f b;
          b.x = sl[brow + c0];
          b.y = sl[brow + c0 + 1];
          acc = __builtin_amdgcn_wmma_f32_16x16x4_f32(false, At[mt * 10 + kc], false, b,
                                                      (short)0, acc, false, false);
        }
        W[mt] = acc;
      }
      // mass_i = sum_d Pprev[i][d] * W[d][i]; D-layout: lane holds N=li, VGPR r -> d = mt*16+half*8+r
      float part = 0.f;
      const int prow = (base + li) * ROWSTR;
#pragma unroll
      for (int mt = 0; mt < 3; ++mt) {
#pragma unroll
        for (int r = 0; r < 8; ++r) {
          int d = mt * 16 + half * 8 + r;
          if (d < C_TAGS) part += W[mt][r] * sl[prow + d];
        }
      }
      part += __shfl_xor(part, 16, 32);
      if (half == 0) {
        int gp = g0 + base + li;                 // global prev-token index
        bool ok = ((gp % T_SEQ) != (T_SEQ - 1))
               && (slab[base + li]     != PAD_IDX)
               && (slab[base + li + 1] != PAD_IDX);
        if (ok) { tr_s += part; tr_c += 1.f; }
      }
    }
  }

  // ---------------- block reduction of the 8 partial quantities ----------------
  float vals[8] = {ce_s, ce_c, ms_s, ms_c, fp_s, fp_c, tr_s, tr_c};
#pragma unroll
  for (int q = 0; q < 8; ++q) {
    float v = vals[q];
#pragma unroll
    for (int o = 16; o > 0; o >>= 1) v += __shfl_xor(v, o, 32);
    if ((tid & 31) == 0) wred[(tid >> 5) * 8 + q] = v;
  }
  __syncthreads();
  if (tid < 8) {
    float s = 0.f;
#pragma unroll
    for (int w = 0; w < 8; ++w) s += wred[w * 8 + tid];
    partials[blockIdx.x * 8 + tid] = s;
  }
}

__global__ __launch_bounds__(256) void ner_loss_finalize(
    const float* __restrict__ partials, int nblocks, float* __restrict__ out)
{
  __shared__ double red[256][8];
  const int tid = threadIdx.x;
  double acc[8] = {0, 0, 0, 0, 0, 0, 0, 0};
  for (int i = tid; i < nblocks; i += 256)
#pragma unroll
    for (int q = 0; q < 8; ++q) acc[q] += (double)partials[i * 8 + q];
#pragma unroll
  for (int q = 0; q < 8; ++q) red[tid][q] = acc[q];
  __syncthreads();
  for (int s = 128; s > 0; s >>= 1) {
    if (tid < s) {
#pragma unroll
      for (int q = 0; q < 8; ++q) red[tid][q] += red[tid + s][q];
    }
    __syncthreads();
  }
  if (tid == 0) {
    double ce = red[0][0] / fmax(red[0][1], 1.0);
    double ms = red[0][2] / fmax(red[0][3], 1.0);
    double fp = red[0][4] / fmax(red[0][5], 1.0);
    double tr = red[0][6] / fmax(red[0][7], 1.0);
    out[0] = (float)(ce + 1.2 * ms + 1.0 * fp + 0.8 * tr);
  }
}

extern "C" void kernel_launch(void* const* d_in, const int* in_sizes, int n_in,
                              void* d_out, int out_size, void* d_ws, size_t ws_size,
                              hipStream_t stream) {
  (void)n_in; (void)out_size; (void)ws_size;
  const float* logits = (const float*)d_in[0];
  const int*   labels = (const int*)d_in[1];
  const float* vmask  = (const float*)d_in[2];
  int total   = in_sizes[1];                 // B*T tokens
  int nblocks = (total + 255) / 256;
  float* partials = (float*)d_ws;            // nblocks*8 floats

  ner_loss_main<<<nblocks, 256, 0, stream>>>(logits, labels, vmask, partials, total);
  ner_loss_finalize<<<1, 256, 0, stream>>>(partials, nblocks, (float*)d_out);
}